// PostProcess_35682588295225
// MI455X (gfx1250) — compile-verified
//
#include <hip/hip_runtime.h>
#include <hip/hip_bf16.h>
#include <stdint.h>

#define C_IN 7
#define H_IN 368
#define W_IN 120
#define OH   736
#define OW   240
#define NB   32
#define BLK  128   // 4 x wave32

// Low 32 bits of a generic pointer to LDS are the LDS byte address
// (flat->LDS aperture mapping truncates to addr[31:0], ISA 10.2).
__device__ __forceinline__ unsigned ldsOffset(const void* p) {
    return (unsigned)(unsigned long long)(uintptr_t)p;
}

// CDNA5 async global->LDS copies (GV mode), tracked by ASYNCcnt.
__device__ __forceinline__ void asyncLoadB32(unsigned ldsoff, const float* g) {
    asm volatile("global_load_async_to_lds_b32 %0, %1, off"
                 :: "v"(ldsoff), "v"(g) : "memory");
}
__device__ __forceinline__ void asyncLoadB64(unsigned ldsoff, const float* g) {
    asm volatile("global_load_async_to_lds_b64 %0, %1, off"
                 :: "v"(ldsoff), "v"(g) : "memory");
}
__device__ __forceinline__ void waitAsync() {
    asm volatile("s_wait_asynccnt 0" ::: "memory");
}

__global__ __launch_bounds__(BLK) void fused_col_kernel(
    const float* __restrict__ x,      // (32,7,368,120)
    const float* __restrict__ dw,     // (6,1,1,2)
    float* __restrict__ out1,         // (32,1,240,6) flattened
    float* __restrict__ cnt_ws)       // (32,240) class-6 counts per column
{
    __shared__ float2 s_x[C_IN * H_IN];      // [c][h] = {x[..,wi0], x[..,wi1]}
    __shared__ int    s_lab[OH];
    __shared__ unsigned long long s_best[6];
    __shared__ float  s_cnt;

    const int w   = blockIdx.x;
    const int b   = blockIdx.y;
    const int tid = threadIdx.x;

    // ---- W-direction align-corners indices (match JAX f32 math) ----
    const float WS = 119.0f / 239.0f;
    float wpos = (float)w * WS;
    int wi0 = (int)floorf(wpos);
    wi0 = wi0 < 119 ? wi0 : 119;
    int wi1 = (wi0 + 1) < 119 ? (wi0 + 1) : 119;
    float wf = wpos - (float)wi0;

    const float* xb    = x + (size_t)b * C_IN * H_IN * W_IN;
    const float* gbase = xb + wi0;           // element e reads gbase + e*W_IN
    unsigned sbase = ldsOffset(&s_x[0]);

    // ---- async-stage both corner columns into LDS ----
    if (wi1 == wi0 + 1) {
        // adjacent input columns: one b64 fetches both corners (all w < 239)
        for (int e = tid; e < C_IN * H_IN; e += BLK)     // e = c*H_IN + h
            asyncLoadB64(sbase + (unsigned)e * 8u, gbase + (size_t)e * W_IN);
    } else {
        // w = 239: wi0 == wi1 == 119, wf == 0 -> duplicate the column
        for (int e = tid; e < C_IN * H_IN; e += BLK) {
            const float* g = gbase + (size_t)e * W_IN;
            asyncLoadB32(sbase + (unsigned)e * 8u,      g);
            asyncLoadB32(sbase + (unsigned)e * 8u + 4u, g);
        }
    }

    if (tid < 6)  s_best[tid] = 735ull;   // key(v=0, h=735)  -> out1 = 0
    if (tid == 6) s_cnt = 0.0f;

    waitAsync();
    __syncthreads();

    // ---- Phase A: bilinear interp + channel argmax -> label per h ----
    const float HS = 367.0f / 735.0f;
    for (int h = tid; h < OH; h += BLK) {
        float hpos = (float)h * HS;
        int hi0 = (int)floorf(hpos);
        hi0 = hi0 < 367 ? hi0 : 367;
        int hi1 = (hi0 + 1) < 367 ? (hi0 + 1) : 367;
        float hf = hpos - (float)hi0;

        float bestv = -3.402823466e38f;
        int bestc = 0;
        #pragma unroll
        for (int c = 0; c < C_IN; ++c) {
            float2 p0 = s_x[c * H_IN + hi0];           // ds_load_b64
            float2 p1 = s_x[c * H_IN + hi1];
            float xh0 = p0.x * (1.0f - hf) + p1.x * hf;  // H interp first
            float xh1 = p0.y * (1.0f - hf) + p1.y * hf;
            float v   = xh0 * (1.0f - wf) + xh1 * wf;    // then W interp
            if (v > bestv) { bestv = v; bestc = c; }     // first-occurrence
        }
        s_lab[h] = bestc;
    }
    __syncthreads();

    // ---- Phase B: per-class flipped-argmax of relu edge + class-6 count ----
    float w0[6], w1[6];
    #pragma unroll
    for (int c = 0; c < 6; ++c) { w0[c] = dw[2 * c]; w1[c] = dw[2 * c + 1]; }

    unsigned long long best[6];
    #pragma unroll
    for (int c = 0; c < 6; ++c) best[c] = 735ull;
    int cnt6 = 0;

    for (int h = tid; h < OH; h += BLK) {
        int labh  = s_lab[h];
        int labh1 = (h < OH - 1) ? s_lab[h + 1] : -1;
        cnt6 += (labh == 6) ? 1 : 0;
        #pragma unroll
        for (int c = 0; c < 6; ++c) {
            float d  = (labh == c) ? 1.0f : 0.0f;
            float pn = (h == OH - 1) ? 0.5f : ((labh1 == c) ? 1.0f : 0.0f);
            float v  = w0[c] * d + w1[c] * pn;
            v = (v > 0.0f) ? v : 0.0f;   // relu; forces +0
            // monotonic key: higher value wins, ties -> larger h
            // (flipped argmax: first occurrence = largest h among maxima)
            unsigned long long key =
                ((unsigned long long)__float_as_uint(v) << 10) | (unsigned)h;
            if (key > best[c]) best[c] = key;
        }
    }
    #pragma unroll
    for (int c = 0; c < 6; ++c) atomicMax(&s_best[c], best[c]);
    atomicAdd(&s_cnt, (float)cnt6);
    __syncthreads();

    if (tid < 6) {
        int hstar = (int)(s_best[tid] & 1023ull);
        out1[((size_t)b * OW + w) * 6 + tid] = (float)(735 - hstar);
    }
    if (tid == 6) cnt_ws[b * OW + w] = s_cnt;
}

__global__ __launch_bounds__(256) void reduce_out2_kernel(
    const float* __restrict__ cnt_ws, float* __restrict__ out2)
{
    int t = threadIdx.x;          // 256 = 32 b x 8 groups
    int b = t >> 3, g = t & 7;
    float s = 0.0f;
    #pragma unroll
    for (int j = 0; j < 30; ++j) s += cnt_ws[b * OW + g * 30 + j];
    out2[t] = s;                   // (B,1,8)
}

extern "C" void kernel_launch(void* const* d_in, const int* in_sizes, int n_in,
                              void* d_out, int out_size, void* d_ws, size_t ws_size,
                              hipStream_t stream) {
    const float* x  = (const float*)d_in[0];   // (32,7,368,120) fp32
    const float* dw = (const float*)d_in[1];   // (6,1,1,2) fp32
    float* out1   = (float*)d_out;                      // 32*240*6 = 46080
    float* out2   = out1 + (size_t)NB * OW * 6;         // 32*8 = 256
    float* cnt_ws = (float*)d_ws;                       // 32*240 floats

    dim3 grid(OW, NB);
    fused_col_kernel<<<grid, BLK, 0, stream>>>(x, dw, out1, cnt_ws);
    reduce_out2_kernel<<<1, 256, 0, stream>>>(cnt_ws, out2);
}